// LBQuantization_35021163331684
// MI455X (gfx1250) — compile-verified
//
#include <hip/hip_runtime.h>
#include <stdint.h>

// Problem constants (fixed by setup_inputs): B=32, c=3, H=W=512
#define NC        96          // B*c channels
#define HW        262144      // H*W elements per channel
#define NSEG      8           // reduction segments per channel
#define RB        256         // reduce block size
#define MB        256         // map block size
#define MAP_SUBS  64          // HW / (MB*16) blocks per channel in map pass

// ws layout (floats):
//   [0      , 768 )  pmin[NC*NSEG]
//   [768    , 1536)  pmax[NC*NSEG]
//   [1536   , 2304)  rec[NC][8] = {min, t0..t6} (sorted)
#define WS_PMIN 0
#define WS_PMAX 768
#define WS_REC  1536

typedef float v4f __attribute__((ext_vector_type(4)));

// ---------------------------------------------------------------------------
// Pass 1: deterministic per-channel-segment min/max partials (no atomics).
// ---------------------------------------------------------------------------
__global__ __launch_bounds__(RB) void reduce_minmax(const float* __restrict__ x,
                                                    float* __restrict__ ws) {
  const int blk = blockIdx.x;          // blk = ch*NSEG + seg
  const int ch  = blk >> 3;
  const int seg = blk & 7;
  // 32768 floats per segment = 8192 float4; 256 threads * 32 iters
  const v4f* p = (const v4f*)x + (size_t)ch * (HW / 4) + seg * 8192 + threadIdx.x;

  float vmin = __builtin_inff();
  float vmax = -__builtin_inff();
#pragma unroll
  for (int i = 0; i < 32; ++i) {
    v4f v = p[i * 256];
    vmin = fminf(vmin, fminf(fminf(v.x, v.y), fminf(v.z, v.w)));
    vmax = fmaxf(vmax, fmaxf(fmaxf(v.x, v.y), fmaxf(v.z, v.w)));
  }
  // wave32 reduction
#pragma unroll
  for (int m = 16; m >= 1; m >>= 1) {
    vmin = fminf(vmin, __shfl_xor(vmin, m, 32));
    vmax = fmaxf(vmax, __shfl_xor(vmax, m, 32));
  }
  __shared__ float smin[RB / 32];
  __shared__ float smax[RB / 32];
  const int wave = threadIdx.x >> 5;
  const int lane = threadIdx.x & 31;
  if (lane == 0) { smin[wave] = vmin; smax[wave] = vmax; }
  __syncthreads();
  if (threadIdx.x == 0) {
    float a = smin[0], b = smax[0];
#pragma unroll
    for (int i = 1; i < RB / 32; ++i) { a = fminf(a, smin[i]); b = fmaxf(b, smax[i]); }
    ws[WS_PMIN + blk] = a;
    ws[WS_PMAX + blk] = b;
  }
}

// ---------------------------------------------------------------------------
// Pass 2: fold partials, compute + sort thresholds, emit {min, t0..t6}/channel.
// No FMA contraction: match JAX's separate mul + add bit-exactly.
// ---------------------------------------------------------------------------
__global__ __launch_bounds__(128) void prep_thresholds(const float* __restrict__ pct,
                                                       float* __restrict__ ws) {
  const int ch = threadIdx.x;
  if (ch >= NC) return;
  float mn = __builtin_inff();
  float mx = -__builtin_inff();
#pragma unroll
  for (int i = 0; i < NSEG; ++i) {
    mn = fminf(mn, ws[WS_PMIN + ch * NSEG + i]);
    mx = fmaxf(mx, ws[WS_PMAX + ch * NSEG + i]);
  }
  const float range = __fsub_rn(mx, mn);
  float t[7];
  for (int j = 0; j < 7; ++j)
    t[j] = __fadd_rn(__fmul_rn(pct[ch * 7 + j], range), mn);
  // insertion sort (7 elements)
  for (int i = 1; i < 7; ++i) {
    float key = t[i];
    int j = i - 1;
    while (j >= 0 && t[j] > key) { t[j + 1] = t[j]; --j; }
    t[j + 1] = key;
  }
  float* rec = ws + WS_REC + ch * 8;
  rec[0] = mn;
  for (int j = 0; j < 7; ++j) rec[1 + j] = t[j];
}

// searchsorted(side='right') ⇒ rid = #{t_j <= v}; proxy = left_ends[rid]
// = largest threshold <= v, else channel min. Monotone select chain.
__device__ __forceinline__ float qsel(float v, const float r[8]) {
  float p = r[0];
  p = (v >= r[1]) ? r[1] : p;
  p = (v >= r[2]) ? r[2] : p;
  p = (v >= r[3]) ? r[3] : p;
  p = (v >= r[4]) ? r[4] : p;
  p = (v >= r[5]) ? r[5] : p;
  p = (v >= r[6]) ? r[6] : p;
  p = (v >= r[7]) ? r[7] : p;
  return p;
}

// ---------------------------------------------------------------------------
// Pass 3: streaming map. x staged through LDS with CDNA5 async global->LDS
// copies (ASYNCcnt); output written with non-temporal stores so the 100 MB
// write stream does not evict x (still warm in the 192 MB L2 from pass 1).
// Each wave waits on its own ASYNCcnt and reads only its own LDS ⇒ no barrier.
// ---------------------------------------------------------------------------
__global__ __launch_bounds__(MB) void map_quant(const float* __restrict__ x,
                                                const float* __restrict__ ws,
                                                float* __restrict__ out) {
  __shared__ v4f tile[4][MB];   // 16 KB
  const int blk = blockIdx.x;   // blk = ch*MAP_SUBS + sub
  const int ch  = blk >> 6;
  const int sub = blk & 63;
  const unsigned tid = threadIdx.x;
  const unsigned base = (unsigned)ch * HW + (unsigned)sub * (MB * 16); // element offset

  const unsigned lds_base = (unsigned)(uintptr_t)&tile[0][0]; // low 32 bits = LDS offset

  // Issue 4 async b128 loads per lane: global (GVS: sgpr base + vgpr byte offset) -> LDS
#pragma unroll
  for (int k = 0; k < 4; ++k) {
    unsigned lds_off = lds_base + (unsigned)((k * MB + tid) * 16);
    unsigned goff    = (base + (unsigned)k * (MB * 4) + tid * 4) * 4u; // bytes
    asm volatile("global_load_async_to_lds_b128 %0, %1, %2"
                 :: "v"(lds_off), "v"(goff), "s"(x)
                 : "memory");
  }

  // Overlap: load the (block-uniform) per-channel record while copies fly.
  const float* recp = ws + WS_REC + ch * 8;
  v4f ra = *(const v4f*)(recp);
  v4f rb = *(const v4f*)(recp + 4);
  float r[8] = {ra.x, ra.y, ra.z, ra.w, rb.x, rb.y, rb.z, rb.w};

  asm volatile("s_wait_asynccnt 0" ::: "memory");

#pragma unroll
  for (int k = 0; k < 4; ++k) {
    v4f v = tile[k][tid];
    v4f o;
    o.x = qsel(v.x, r);
    o.y = qsel(v.y, r);
    o.z = qsel(v.z, r);
    o.w = qsel(v.w, r);
    v4f* dst = (v4f*)(out + base) + k * MB + tid;
    __builtin_nontemporal_store(o, dst);
  }
}

// ---------------------------------------------------------------------------
extern "C" void kernel_launch(void* const* d_in, const int* in_sizes, int n_in,
                              void* d_out, int out_size, void* d_ws, size_t ws_size,
                              hipStream_t stream) {
  const float* x   = (const float*)d_in[0];  // (32,3,512,512) f32
  const float* pct = (const float*)d_in[1];  // (96,7) f32
  float* out = (float*)d_out;                // (32,3,512,512) f32
  float* ws  = (float*)d_ws;                 // needs 2304 floats (~9.2 KB)

  reduce_minmax  <<<NC * NSEG,       RB, 0, stream>>>(x, ws);
  prep_thresholds<<<1,              128, 0, stream>>>(pct, ws);
  map_quant      <<<NC * MAP_SUBS,   MB, 0, stream>>>(x, ws, out);
}